// DetectionTargetLayer_29094108463155
// MI455X (gfx1250) — compile-verified
//
#include <hip/hip_runtime.h>
#include <hip/hip_bf16.h>
#include <math.h>

// ---------------------------------------------------------------------------
// DetectionTargetLayer for MI455X (gfx1250, wave32).
//
// Two fused kernels:
//   dtl_select : 1 block / image (1024 thr = 32 waves). GT boxes in LDS,
//                IoU + crowd masks, wave32 ballot-scan stable compaction,
//                deltas + class ids + roi outputs, positive-ROI records to ws.
//   dtl_masks  : 1 block / output ROI slot (B*200 blocks, 256 thr). Bilinear
//                crop-resize of the matched GT mask via 4-tap byte gathers
//                issued as global_load_async_to_lds_b8 (ASYNCcnt path),
//                s_wait_asynccnt 0, then per-lane LDS reads.
// ---------------------------------------------------------------------------

static constexpr int NPROP = 2000;   // proposals per image
static constexpr int GMAX  = 100;    // GT slots per image
static constexpr int PCAP  = 66;     // positive ROI cap
static constexpr int NCAP  = 134;    // negative ROI cap
static constexpr int TROIS = 200;    // total ROIs out per image
static constexpr int MH    = 1024;   // mask height
static constexpr int MW    = 1024;   // mask width
static constexpr int SOUT  = 28;     // output mask side

// ---------------------------------------------------------------------------
// Kernel A: per-image selection / compaction / deltas.
// ---------------------------------------------------------------------------
__global__ __launch_bounds__(1024) void dtl_select(
    const float* __restrict__ proposals,   // [B, NPROP, 4]
    const int*   __restrict__ gt_cls,      // [B, GMAX]
    const float* __restrict__ gt_boxes,    // [B, GMAX, 4]
    float* __restrict__ out_rois,          // [B, TROIS, 4]
    int*   __restrict__ out_cls,           // [B, TROIS]
    float* __restrict__ out_deltas,        // [B, TROIS, 4]
    float* __restrict__ wsf)               // [B, PCAP, 8] pos-ROI records
{
  const int b = blockIdx.x;
  const int t = threadIdx.x;

  __shared__ float sgt[GMAX * 4];
  __shared__ int   scls[GMAX];
  __shared__ int   posIdx[PCAP];
  __shared__ int   negIdx[NCAP];
  __shared__ int   swtP[32], swtN[32];
  __shared__ int   basePos, baseNeg;

  if (t < GMAX) {
    float4 gb = reinterpret_cast<const float4*>(gt_boxes)[b * GMAX + t];
    sgt[t * 4 + 0] = gb.x; sgt[t * 4 + 1] = gb.y;
    sgt[t * 4 + 2] = gb.z; sgt[t * 4 + 3] = gb.w;
    scls[t] = gt_cls[b * GMAX + t];
  }
  if (t < PCAP) posIdx[t] = -1;
  if (t < NCAP) negIdx[t] = -1;
  if (t == 0) { basePos = 0; baseNeg = 0; }
  __syncthreads();

  const int lane = t & 31;
  const int wv   = t >> 5;

  // --- classify every proposal, stable-compact positives & negatives -------
  for (int c = 0; c < (NPROP + 1023) / 1024; ++c) {
    const int i = c * 1024 + t;
    bool pos = false, neg = false;
    if (i < NPROP) {
      __builtin_prefetch(proposals + ((size_t)b * NPROP + i) * 4);  // global_prefetch_b8
      float4 pr = reinterpret_cast<const float4*>(proposals)[b * NPROP + i];
      const bool pvalid = (pr.x != 0.f) | (pr.y != 0.f) | (pr.z != 0.f) | (pr.w != 0.f);
      const float a1 = (pr.z - pr.x) * (pr.w - pr.y);
      float ovmax = -1.f, crmax = -1.f;
      for (int g = 0; g < GMAX; ++g) {
        const float g0 = sgt[g * 4 + 0], g1 = sgt[g * 4 + 1];
        const float g2 = sgt[g * 4 + 2], g3 = sgt[g * 4 + 3];
        const int   cl = scls[g];
        const bool gval = (g0 != 0.f) | (g1 != 0.f) | (g2 != 0.f) | (g3 != 0.f);
        const float yy1 = fmaxf(pr.x, g0), xx1 = fmaxf(pr.y, g1);
        const float yy2 = fminf(pr.z, g2), xx2 = fminf(pr.w, g3);
        const float inter = fmaxf(yy2 - yy1, 0.f) * fmaxf(xx2 - xx1, 0.f);
        const float a2 = (g2 - g0) * (g3 - g1);
        const float un = a1 + a2 - inter;
        const float iou = inter / (un > 0.f ? un : 1.f);
        if (gval && cl > 0) ovmax = fmaxf(ovmax, iou);
        if (gval && cl < 0) crmax = fmaxf(crmax, iou);
      }
      pos = (ovmax >= 0.5f) && pvalid;
      neg = (ovmax <  0.5f) && (crmax < 0.001f) && pvalid;
    }

    const unsigned bp = (unsigned)__ballot(pos);   // wave32: low 32 bits
    const unsigned bn = (unsigned)__ballot(neg);
    if (lane == 0) { swtP[wv] = __popc(bp); swtN[wv] = __popc(bn); }
    __syncthreads();

    const int prefP = __builtin_amdgcn_mbcnt_lo(bp, 0);
    const int prefN = __builtin_amdgcn_mbcnt_lo(bn, 0);
    int sumP = 0, sumN = 0, totP = 0, totN = 0;
    for (int w = 0; w < 32; ++w) {
      const int vp = swtP[w], vn = swtN[w];
      if (w < wv) { sumP += vp; sumN += vn; }
      totP += vp; totN += vn;
    }
    const int rp = basePos + sumP + prefP;
    const int rn = baseNeg + sumN + prefN;
    if (pos && rp < PCAP) posIdx[rp] = i;
    if (neg && rn < NCAP) negIdx[rn] = i;
    __syncthreads();
    if (t == 0) { basePos += totP; baseNeg += totN; }
    __syncthreads();
  }

  const int num_pos    = basePos < PCAP ? basePos : PCAP;
  const int negFound   = baseNeg < NCAP ? baseNeg : NCAP;
  const int neg_target = (int)((float)num_pos / 0.33f) - num_pos;  // f32 like ref

  // --- positive slots: match GT, deltas, outputs, ws record ----------------
  if (t < PCAP) {
    const int k = t;
    const bool ok = (k < num_pos) && (posIdx[k] >= 0);
    float r0 = 0, r1 = 0, r2 = 0, r3 = 0;
    float d0 = 0, d1 = 0, d2 = 0, d3 = 0;
    int clsv = 0; float argf = 0.f;
    if (ok) {
      const int i = posIdx[k];
      float4 pr = reinterpret_cast<const float4*>(proposals)[b * NPROP + i];
      r0 = pr.x; r1 = pr.y; r2 = pr.z; r3 = pr.w;
      const float a1 = (r2 - r0) * (r3 - r1);
      float best = -2.f; int arg = 0;
      for (int g = 0; g < GMAX; ++g) {          // first-occurrence argmax (jnp.argmax)
        const float g0 = sgt[g * 4 + 0], g1 = sgt[g * 4 + 1];
        const float g2 = sgt[g * 4 + 2], g3 = sgt[g * 4 + 3];
        const int   cl = scls[g];
        const bool gok = ((g0 != 0.f) | (g1 != 0.f) | (g2 != 0.f) | (g3 != 0.f)) && (cl > 0);
        const float yy1 = fmaxf(r0, g0), xx1 = fmaxf(r1, g1);
        const float yy2 = fminf(r2, g2), xx2 = fminf(r3, g3);
        const float inter = fmaxf(yy2 - yy1, 0.f) * fmaxf(xx2 - xx1, 0.f);
        const float a2 = (g2 - g0) * (g3 - g1);
        const float un = a1 + a2 - inter;
        const float v  = gok ? (inter / (un > 0.f ? un : 1.f)) : -1.f;
        if (v > best) { best = v; arg = g; }
      }
      const float g0 = sgt[arg * 4 + 0], g1 = sgt[arg * 4 + 1];
      const float g2 = sgt[arg * 4 + 2], g3 = sgt[arg * 4 + 3];
      clsv = scls[arg];
      const float h  = r2 - r0, w_ = r3 - r1;
      const float gh = g2 - g0, gw = g3 - g1;
      const float cy  = r0 + 0.5f * h,  cx  = r1 + 0.5f * w_;
      const float gcy = g0 + 0.5f * gh, gcx = g1 + 0.5f * gw;
      d0 = ((gcy - cy) / h)  / 0.1f;
      d1 = ((gcx - cx) / w_) / 0.1f;
      d2 = logf(gh / h)      / 0.2f;
      d3 = logf(gw / w_)     / 0.2f;
      argf = (float)arg;
    }
    const int row = b * TROIS + k;
    out_rois[row * 4 + 0] = r0; out_rois[row * 4 + 1] = r1;
    out_rois[row * 4 + 2] = r2; out_rois[row * 4 + 3] = r3;
    out_cls[row] = clsv;
    out_deltas[row * 4 + 0] = d0; out_deltas[row * 4 + 1] = d1;
    out_deltas[row * 4 + 2] = d2; out_deltas[row * 4 + 3] = d3;
    float* w8 = wsf + ((size_t)b * PCAP + k) * 8;
    w8[0] = r0; w8[1] = r1; w8[2] = r2; w8[3] = r3;
    w8[4] = argf; w8[5] = ok ? 1.f : 0.f; w8[6] = 0.f; w8[7] = 0.f;
  } else if (t < TROIS) {
    // --- negative slots ----------------------------------------------------
    const int k = t - PCAP;
    const bool ok = (k < negFound) && (k < neg_target) && (negIdx[k] >= 0);
    float r0 = 0, r1 = 0, r2 = 0, r3 = 0;
    if (ok) {
      float4 pr = reinterpret_cast<const float4*>(proposals)[b * NPROP + negIdx[k]];
      r0 = pr.x; r1 = pr.y; r2 = pr.z; r3 = pr.w;
    }
    const int row = b * TROIS + PCAP + k;
    out_rois[row * 4 + 0] = r0; out_rois[row * 4 + 1] = r1;
    out_rois[row * 4 + 2] = r2; out_rois[row * 4 + 3] = r3;
    out_cls[row] = 0;
    out_deltas[row * 4 + 0] = 0.f; out_deltas[row * 4 + 1] = 0.f;
    out_deltas[row * 4 + 2] = 0.f; out_deltas[row * 4 + 3] = 0.f;
  }
}

// ---------------------------------------------------------------------------
// Kernel B: mask crop_and_resize. One block per (image, roi-slot).
// 4-tap byte gathers go through the CDNA5 async global->LDS path.
// ---------------------------------------------------------------------------
__global__ __launch_bounds__(256) void dtl_masks(
    const unsigned char* __restrict__ gm,   // [B, MH, MW, GMAX] bool bytes
    const float* __restrict__ wsf,          // [B, PCAP, 8]
    float* __restrict__ out_masks)          // [B, TROIS, 28*28]
{
  const int blk = blockIdx.x;
  const int b = blk / TROIS;
  const int s = blk % TROIS;
  const int t = threadIdx.x;
  float* mout = out_masks + (size_t)(b * TROIS + s) * (SOUT * SOUT);

  bool  compute = false;
  float y1v = 0, x1v = 0, y2v = 0, x2v = 0;
  int   g = 0;
  if (s < PCAP) {
    const float* w8 = wsf + ((size_t)b * PCAP + s) * 8;
    if (w8[5] != 0.f) {
      compute = true;
      y1v = w8[0]; x1v = w8[1]; y2v = w8[2]; x2v = w8[3];
      g = (int)w8[4];
    }
  }
  if (!compute) {  // uniform per block
    for (int p = t; p < SOUT * SOUT; p += 256) mout[p] = 0.f;
    return;
  }

  __shared__ unsigned char lbuf[256 * 4];
  // Flat pointer to LDS: low 32 bits are the LDS byte offset (ISA 10.2).
  const unsigned lbase = (unsigned)(unsigned long long)(uintptr_t)lbuf;
  const size_t ibase = (size_t)b * ((size_t)MH * MW * GMAX) + (unsigned)g;

  for (int k = 0; k < 4; ++k) {
    const int p  = t + k * 256;
    const int pc = p < SOUT * SOUT ? p : SOUT * SOUT - 1;
    const int i = pc / SOUT, j = pc % SOUT;

    const float fy = (y1v + (y2v - y1v) * ((float)i / 27.0f)) * 1023.0f;
    const float fx = (x1v + (x2v - x1v) * ((float)j / 27.0f)) * 1023.0f;
    const float y0f = fminf(fmaxf(floorf(fy), 0.f), 1023.f);
    const float x0f = fminf(fmaxf(floorf(fx), 0.f), 1023.f);
    const int y0 = (int)y0f, x0 = (int)x0f;
    const int y1i = (y0 + 1) < 1023 ? (y0 + 1) : 1023;
    const int x1i = (x0 + 1) < 1023 ? (x0 + 1) : 1023;
    const float wy = fy - y0f, wx = fx - x0f;

    const unsigned long long a00 = (unsigned long long)(uintptr_t)(gm + ibase + ((size_t)y0  * MW + x0 ) * GMAX);
    const unsigned long long a01 = (unsigned long long)(uintptr_t)(gm + ibase + ((size_t)y0  * MW + x1i) * GMAX);
    const unsigned long long a10 = (unsigned long long)(uintptr_t)(gm + ibase + ((size_t)y1i * MW + x0 ) * GMAX);
    const unsigned long long a11 = (unsigned long long)(uintptr_t)(gm + ibase + ((size_t)y1i * MW + x1i) * GMAX);
    const unsigned l0 = lbase + (unsigned)t * 4u;

    // CDNA5 async global->LDS byte gathers (ASYNCcnt), then wait + LDS read.
    asm volatile("global_load_async_to_lds_b8 %0, %1, off" :: "v"(l0),      "v"(a00) : "memory");
    asm volatile("global_load_async_to_lds_b8 %0, %1, off" :: "v"(l0 + 1u), "v"(a01) : "memory");
    asm volatile("global_load_async_to_lds_b8 %0, %1, off" :: "v"(l0 + 2u), "v"(a10) : "memory");
    asm volatile("global_load_async_to_lds_b8 %0, %1, off" :: "v"(l0 + 3u), "v"(a11) : "memory");
    asm volatile("s_wait_asynccnt 0" ::: "memory");

    const float m00 = (float)lbuf[t * 4 + 0];
    const float m01 = (float)lbuf[t * 4 + 1];
    const float m10 = (float)lbuf[t * 4 + 2];
    const float m11 = (float)lbuf[t * 4 + 3];

    const float v = m00 * (1.f - wy) * (1.f - wx) + m01 * (1.f - wy) * wx
                  + m10 * wy * (1.f - wx)        + m11 * wy * wx;
    if (p < SOUT * SOUT) mout[p] = rintf(v);   // round-half-even == jnp.round
  }
}

// ---------------------------------------------------------------------------
extern "C" void kernel_launch(void* const* d_in, const int* in_sizes, int n_in,
                              void* d_out, int out_size, void* d_ws, size_t ws_size,
                              hipStream_t stream)
{
  (void)n_in; (void)out_size; (void)ws_size;
  const float*         proposals = (const float*)d_in[0];
  const int*           gt_cls    = (const int*)d_in[1];
  const float*         gt_boxes  = (const float*)d_in[2];
  const unsigned char* gt_masks  = (const unsigned char*)d_in[3];

  const int B = in_sizes[1] / GMAX;   // batch size (4)

  float* out        = (float*)d_out;
  float* out_rois   = out;                                           // [B,200,4]
  int*   out_cls    = (int*)(out + (size_t)B * TROIS * 4);           // [B,200] int32
  float* out_deltas = out + (size_t)B * TROIS * 4 + (size_t)B * TROIS;  // [B,200,4]
  float* out_masks  = out_deltas + (size_t)B * TROIS * 4;            // [B,200,28,28]
  float* wsf        = (float*)d_ws;                                  // [B,66,8]

  dtl_select<<<B, 1024, 0, stream>>>(proposals, gt_cls, gt_boxes,
                                     out_rois, out_cls, out_deltas, wsf);
  dtl_masks<<<B * TROIS, 256, 0, stream>>>(gt_masks, wsf, out_masks);
}